// RAVENexp_74698071212082
// MI455X (gfx1250) — compile-verified
//
#include <hip/hip_runtime.h>
#include <hip/hip_bf16.h>

// RAVEN quantized-Taylor exp: out = exp(floor(x)) * sum_i fxp(coeff[i]*frac^i)
// Pure elementwise stream: 537 MB @ 23.3 TB/s -> ~23 us floor; ~27 VALU ops/elem
// (much of it VOPD dual-issuable) keeps compute at/under the memory time.
// fp32 VALU path (the rint/clamp grid must be exact), NT b128 loads/stores
// (537 MB stream > 192 MB L2, zero reuse), no contraction -> WMMA inapplicable.

typedef float v4f __attribute__((ext_vector_type(4)));

#define THREADS 256
#define VPT 4      // float4 vectors per thread (16 elements/thread)
#define NCYC 4     // Taylor terms

struct FxpConst {
    float cs[NCYC];     // coeff[i] * 2^fw[i]   (exact: power-of-two scale)
    float inv_s[NCYC];  // 2^-fw[i]             (exact, via exp2f(-fw): no division)
    float ghi[NCYC];    // 2^(iw+fw) - 1        (grid-domain clamp hi, exact)
    float glo[NCYC];    // -2^(iw+fw)           (grid-domain clamp lo, exact)
};

// Bit-exact vs reference:
//   ref: term = clip(rint(c*vp*2^fw)/2^fw, -2^iw, 2^iw-2^-fw); acc += term
//   here: r = clamp_grid(rint(cs*vp));  acc = fma(r, 2^-fw, acc)
// Power-of-two scaling is exact & monotone -> grid-domain clamp == ref clamp;
// r*2^-fw is exact -> the FMA's single rounding == the ref add's rounding.
__device__ __forceinline__ float raven_one(float xv, const FxpConst& k, float term0) {
    float ip    = floorf(xv);
    float f     = xv - ip;          // frac in [0,1)
    float scale = expf(ip);         // exp of the integer part
    float acc   = term0;            // i=0 term: vp==1 -> thread-uniform, hoisted
    float vp    = f;
#pragma unroll
    for (int i = 1; i < NCYC; ++i) {
        float r = rintf(k.cs[i] * vp);               // round-half-to-even on grid
        r = fminf(fmaxf(r, k.glo[i]), k.ghi[i]);     // clamp on grid (exact)
        acc = fmaf(r, k.inv_s[i], acc);              // exact rescale fused into add
        vp *= f;                                     // f, f^2, f^3 (last mul is dead)
    }
    return acc * scale;
}

__global__ __launch_bounds__(THREADS) void raven_exp_kernel(
    const float* __restrict__ x,
    const float* __restrict__ coeff,
    const float* __restrict__ intw,
    const float* __restrict__ fracw,
    float* __restrict__ out,
    int n)
{
    // Thread-uniform fixed-point constants (scalar loads; exp2f of uniform
    // values scalarizes onto the SALU float pipe; no divisions anywhere).
    FxpConst k;
#pragma unroll
    for (int i = 0; i < NCYC; ++i) {
        k.cs[i]    = coeff[i] * exp2f(fracw[i]);   // exact scaling of coeff
        k.inv_s[i] = exp2f(-fracw[i]);             // exact 2^-fw, division-free
        float g    = exp2f(intw[i] + fracw[i]);
        k.ghi[i]   = g - 1.0f;                     // (2^iw - 2^-fw)*2^fw, exact
        k.glo[i]   = -g;
    }
    // Term 0 is data-independent (vp == 1): fully uniform, computed once.
    const float term0 =
        fminf(fmaxf(rintf(k.cs[0]), k.glo[0]), k.ghi[0]) * k.inv_s[0];

    const int n4 = n >> 2;
    const v4f* __restrict__ xv = (const v4f*)x;
    v4f* __restrict__ ov       = (v4f*)out;

    const int tile = THREADS * VPT;
    const int base = blockIdx.x * tile;

    if (base + tile <= n4) {
        // Fast path (uniform branch, >99.9% of blocks): unconditional
        // NT b128 loads (4 independent in flight), compute, NT stores.
        const int v0 = base + threadIdx.x;
        v4f in[VPT];
#pragma unroll
        for (int j = 0; j < VPT; ++j)
            in[j] = __builtin_nontemporal_load(&xv[v0 + j * THREADS]);
#pragma unroll
        for (int j = 0; j < VPT; ++j) {
            v4f r;
#pragma unroll
            for (int e = 0; e < 4; ++e) r[e] = raven_one(in[j][e], k, term0);
            __builtin_nontemporal_store(r, &ov[v0 + j * THREADS]);
        }
    } else {
        // Edge block: per-vector guards.
#pragma unroll
        for (int j = 0; j < VPT; ++j) {
            int v = base + threadIdx.x + j * THREADS;
            if (v < n4) {
                v4f d = __builtin_nontemporal_load(&xv[v]);
                v4f r;
#pragma unroll
                for (int e = 0; e < 4; ++e) r[e] = raven_one(d[e], k, term0);
                __builtin_nontemporal_store(r, &ov[v]);
            }
        }
    }

    // Scalar tail for n % 4 (empty for this problem size, kept for generality).
    const int tail = n & 3;
    if (tail && blockIdx.x == 0 && (int)threadIdx.x < tail) {
        int e = (n & ~3) + threadIdx.x;
        out[e] = raven_one(x[e], k, term0);
    }
}

extern "C" void kernel_launch(void* const* d_in, const int* in_sizes, int n_in,
                              void* d_out, int out_size, void* d_ws, size_t ws_size,
                              hipStream_t stream) {
    const float* x     = (const float*)d_in[0];
    const float* coeff = (const float*)d_in[1];
    const float* intw  = (const float*)d_in[2];
    const float* fracw = (const float*)d_in[3];
    float*       out   = (float*)d_out;

    const int n         = in_sizes[0];
    const int n4        = n >> 2;
    const int per_block = THREADS * VPT;
    int blocks = (n4 + per_block - 1) / per_block;
    if (blocks < 1) blocks = 1;

    raven_exp_kernel<<<blocks, THREADS, 0, stream>>>(x, coeff, intw, fracw, out, n);
}